// small_GCN_14774687498697
// MI455X (gfx1250) — compile-verified
//
#include <hip/hip_runtime.h>
#include <hip/hip_bf16.h>

typedef __attribute__((ext_vector_type(2))) float v2f;
typedef __attribute__((ext_vector_type(8))) float v8f;

#define F_IN    128
#define HIDDEN  64
#define N_GRAPHS 64
#define N_CLASSES 10

// ---------------------------------------------------------------------------
// 0) workspace init: deg = 1.0 (self-loop), h/pooled/counts = 0
// ---------------------------------------------------------------------------
__global__ void gcn_init_ws(float* __restrict__ deg, float* __restrict__ zero_region,
                            long long zero_count, int n_nodes) {
    long long gid    = (long long)blockIdx.x * blockDim.x + threadIdx.x;
    long long stride = (long long)gridDim.x * blockDim.x;
    for (long long i = gid; i < zero_count; i += stride) zero_region[i] = 0.0f;
    for (long long i = gid; i < n_nodes;   i += stride) deg[i] = 1.0f;
}

// ---------------------------------------------------------------------------
// 1) degree accumulation over edge destinations
// ---------------------------------------------------------------------------
__global__ void gcn_degree(const int* __restrict__ dst, float* __restrict__ deg, int n_edges) {
    int e = blockIdx.x * blockDim.x + threadIdx.x;
    if (e < n_edges) atomicAdd(&deg[dst[e]], 1.0f);
}

// 2) dinv = rsqrt(deg)   (deg >= 1 always, self-loop included)
__global__ void gcn_rsqrt(float* __restrict__ deg, int n_nodes) {
    int n = blockIdx.x * blockDim.x + threadIdx.x;
    if (n < n_nodes) deg[n] = rsqrtf(deg[n]);
}

// ---------------------------------------------------------------------------
// 3) xw = x @ W   via V_WMMA_F32_16X16X4_F32, one 16x16 tile per wave32
//    A 16x4 f32: lane L -> row L%16; VGPR0/1 hold K = {0,1} (lanes 0-15)
//                or K = {2,3} (lanes 16-31)
//    B 4x16 f32: lane L -> col L%16; lanes 0-15 hold K = {0,1}, lanes 16-31
//                hold K = {2,3}
//    C/D 16x16 f32: VGPR i -> row i + 8*(lane>=16), col = lane%16
// ---------------------------------------------------------------------------
__global__ void gcn_gemm_wmma(const float* __restrict__ x, const float* __restrict__ W,
                              float* __restrict__ xw, int n_nodes) {
    const int wave = threadIdx.x >> 5;           // 8 waves / block
    const int lane = threadIdx.x & 31;
    const int nRowTiles = (n_nodes + 15) >> 4;
    const int tileId = blockIdx.x * 8 + wave;
    if (tileId >= nRowTiles * 4) return;         // wave-uniform exit
    const int rowTile = tileId >> 2;
    const int colTile = tileId & 3;
    const int rowBase = rowTile << 4;
    const int colBase = colTile << 4;
    const int half = lane >> 4;                  // 0: lanes 0-15, 1: lanes 16-31
    const int l16  = lane & 15;

    if (rowBase + 16 <= n_nodes) {               // full tile: WMMA (EXEC all 1s)
        const float* xrow = x + (long long)(rowBase + l16) * F_IN;
        const float* wcol = W + colBase + l16;
        v8f acc = {};
#pragma unroll
        for (int k = 0; k < F_IN; k += 4) {
            const int kk = k + (half << 1);      // this half's K pair
            v2f a, b;
            a.x = xrow[kk];
            a.y = xrow[kk + 1];
            b.x = wcol[(long long)kk * HIDDEN];
            b.y = wcol[(long long)(kk + 1) * HIDDEN];
            acc = __builtin_amdgcn_wmma_f32_16x16x4_f32(
                false, a, false, b, (short)0, acc, false, false);
        }
#pragma unroll
        for (int i = 0; i < 8; ++i) {
            const int row = rowBase + i + (half << 3);
            xw[(long long)row * HIDDEN + colBase + l16] = acc[i];
        }
    } else {                                     // scalar tail (N%16 != 0 only)
        const int row = rowBase + l16;
        if (half == 0 && row < n_nodes) {
            for (int c = 0; c < 16; ++c) {
                float s = 0.0f;
                for (int k = 0; k < F_IN; ++k)
                    s += x[(long long)row * F_IN + k] * W[(long long)k * HIDDEN + colBase + c];
                xw[(long long)row * HIDDEN + colBase + c] = s;
            }
        }
    }
}

// ---------------------------------------------------------------------------
// 4) edge scatter: h[dst] += xw[src] * dinv[src]*dinv[dst]
//    one thread per (edge, feature); xw/h are L2-resident (25.6 MB each)
// ---------------------------------------------------------------------------
__global__ void gcn_scatter(const int* __restrict__ src, const int* __restrict__ dst,
                            const float* __restrict__ dinv, const float* __restrict__ xw,
                            float* __restrict__ h, int n_edges) {
    long long gid = (long long)blockIdx.x * blockDim.x + threadIdx.x;
    long long total = (long long)n_edges * HIDDEN;
    if (gid >= total) return;
    const int e = (int)(gid >> 6);
    const int f = (int)(gid & (HIDDEN - 1));
    const int s = __ldg(&src[e]);
    const int d = __ldg(&dst[e]);
    const float norm = __ldg(&dinv[s]) * __ldg(&dinv[d]);
    atomicAdd(&h[(long long)d * HIDDEN + f], __ldg(&xw[(long long)s * HIDDEN + f]) * norm);
}

// ---------------------------------------------------------------------------
// 5) self-loop + bias + global mean-pool accumulation
// ---------------------------------------------------------------------------
__global__ void gcn_pool(const float* __restrict__ h, const float* __restrict__ xw,
                         const float* __restrict__ dinv, const float* __restrict__ bias,
                         const int* __restrict__ batch, float* __restrict__ pooled,
                         float* __restrict__ counts, int n_nodes) {
    long long gid = (long long)blockIdx.x * blockDim.x + threadIdx.x;
    long long total = (long long)n_nodes * HIDDEN;
    if (gid >= total) return;
    const int n = (int)(gid >> 6);
    const int f = (int)(gid & (HIDDEN - 1));
    const float di = dinv[n];
    const float hv = h[gid] + xw[gid] * di * di + bias[f];
    const int g = batch[n];
    atomicAdd(&pooled[g * HIDDEN + f], hv);
    if (f == 0) atomicAdd(&counts[g], 1.0f);
}

// ---------------------------------------------------------------------------
// 6) out[g,c] = (pooled[g]/count[g]) . lin_W[c] + lin_b[c]   (64x10, tiny)
// ---------------------------------------------------------------------------
__global__ void gcn_final(const float* __restrict__ pooled, const float* __restrict__ counts,
                          const float* __restrict__ linW, const float* __restrict__ linb,
                          float* __restrict__ out) {
    int t = blockIdx.x * blockDim.x + threadIdx.x;
    if (t >= N_GRAPHS * N_CLASSES) return;
    const int g = t / N_CLASSES;
    const int c = t % N_CLASSES;
    const float inv = 1.0f / fmaxf(counts[g], 1.0f);
    float sum = linb[c];
#pragma unroll
    for (int f = 0; f < HIDDEN; ++f)
        sum += pooled[g * HIDDEN + f] * inv * linW[c * HIDDEN + f];
    out[t] = sum;
}

extern "C" void kernel_launch(void* const* d_in, const int* in_sizes, int n_in,
                              void* d_out, int out_size, void* d_ws, size_t ws_size,
                              hipStream_t stream) {
    const float* x     = (const float*)d_in[0];
    const int*   edge  = (const int*)d_in[1];
    const int*   batch = (const int*)d_in[2];
    const float* W     = (const float*)d_in[3];
    const float* bias  = (const float*)d_in[4];
    const float* linW  = (const float*)d_in[5];
    const float* linb  = (const float*)d_in[6];
    float* out = (float*)d_out;

    const int N = in_sizes[2];          // 100000
    const int E = in_sizes[1] / 2;      // 3200000
    const int* src = edge;              // edge_index[0]
    const int* dst = edge + E;          // edge_index[1]

    // workspace layout (fp32)
    float* ws     = (float*)d_ws;
    float* deg    = ws;                                    // [N] -> becomes dinv
    float* xw     = ws + (((long long)N + 255) & ~255LL);  // [N*64]
    float* h      = xw + (long long)N * HIDDEN;            // [N*64]
    float* pooled = h + (long long)N * HIDDEN;             // [64*64]
    float* counts = pooled + N_GRAPHS * HIDDEN;            // [64]
    const long long zero_count = (long long)N * HIDDEN + N_GRAPHS * HIDDEN + N_GRAPHS;

    // 0) init
    gcn_init_ws<<<2048, 256, 0, stream>>>(deg, h, zero_count, N);
    // 1) degree
    gcn_degree<<<(E + 255) / 256, 256, 0, stream>>>(dst, deg, E);
    // 2) dinv
    gcn_rsqrt<<<(N + 255) / 256, 256, 0, stream>>>(deg, N);
    // 3) xw = x @ W via WMMA: 4 col tiles x ceil(N/16) row tiles, 8 waves/block
    {
        const int nRowTiles = (N + 15) / 16;
        const int totalTiles = nRowTiles * 4;
        gcn_gemm_wmma<<<(totalTiles + 7) / 8, 256, 0, stream>>>(x, W, xw, N);
    }
    // 4) edge scatter
    {
        const long long total = (long long)E * HIDDEN;
        gcn_scatter<<<(unsigned)((total + 255) / 256), 256, 0, stream>>>(src, dst, deg, xw, h, E);
    }
    // 5) self-loop + bias + pool
    {
        const long long total = (long long)N * HIDDEN;
        gcn_pool<<<(unsigned)((total + 255) / 256), 256, 0, stream>>>(h, xw, deg, bias, batch,
                                                                     pooled, counts, N);
    }
    // 6) final linear
    gcn_final<<<(N_GRAPHS * N_CLASSES + 255) / 256, 256, 0, stream>>>(pooled, counts, linW, linb, out);
}